// HQNN_2628519985614
// MI455X (gfx1250) — compile-verified
//
#include <hip/hip_runtime.h>
#include <math.h>

typedef __attribute__((ext_vector_type(2))) float v2f;
typedef __attribute__((ext_vector_type(8))) float v8f;

#define WAVES_PER_BLOCK 8
#define THREADS (WAVES_PER_BLOCK * 32)

__device__ __forceinline__ float shflx(float v, int mask) {
  return __shfl_xor(v, mask, 32);
}

// branchless tanh: copysign(1 - 2/(e^{2|x|}+1), x); exp overflow saturates to 1
__device__ __forceinline__ float tanh_fast(float v) {
  float e = __expf(2.f * fabsf(v));
  float t = 1.f - 2.f * __builtin_amdgcn_rcpf(e + 1.f);
  return copysignf(t, v);
}

// D = A(16x4) * B(4x16) + C  ; f32 WMMA
__device__ __forceinline__ v8f wmma4(v2f a, v2f b, v8f c) {
  return __builtin_amdgcn_wmma_f32_16x16x4_f32(false, a, false, b, (short)0, c,
                                               false, false);
}

__global__ __launch_bounds__(THREADS) void hqnn_kernel(
    const float* __restrict__ x,      // (B,16)
    const float* __restrict__ theta,  // (3,2,4)
    const float* __restrict__ W0, const float* __restrict__ b0,
    const float* __restrict__ W1, const float* __restrict__ b1,
    const float* __restrict__ W2, const float* __restrict__ b2,
    float* __restrict__ out,          // (B,4)
    int B) {
  __shared__ float lds_UT[3][32 * 32];               // U_i^T, 12 KB
  __shared__ float lds_Wp[24 * 16];                  // zero-padded [W0p|W1p|W2p]
  __shared__ float lds_bp[3 * 16];                   // zero-padded biases
  __shared__ float lds_h[WAVES_PER_BLOCK][16 * 16];  // angles, 16 rows x 16 cols
  __shared__ v2f   lds_trig[WAVES_PER_BLOCK][64];    // (cos,sin) per (wire,row)
  __shared__ float lds_S[WAVES_PER_BLOCK][16 * 32];  // 16 rows x [re|im]

  const int tid  = threadIdx.x;
  const int lane = tid & 31;
  const int wv   = tid >> 5;
  const int n16  = lane & 15;  // WMMA column / A-fragment row
  const int hh   = lane >> 4;  // wave half (selects K sub-pair)

  // -------- block setup: zero-padded weights (unconditional B-frag loads) ----
  for (int idx = tid; idx < 24 * 16; idx += THREADS) {
    const int n = idx & 15, k = idx >> 4;  // k: 0..15 W0 | 16..19 W1 | 20..23 W2
    float v = 0.f;
    if (n < 4) {
      if (k < 16)      v = W0[k * 4 + n];
      else if (k < 20) v = W1[(k - 16) * 4 + n];
      else             v = W2[(k - 20) * 4 + n];
    }
    lds_Wp[idx] = v;
  }
  for (int idx = tid; idx < 48; idx += THREADS) {
    const int n = idx & 15, li = idx >> 4;
    const float* bs = (li == 0) ? b0 : (li == 1) ? b1 : b2;
    lds_bp[idx] = (n < 4) ? bs[n] : 0.f;
  }

  // -------- block setup: build U_i^T (wave 0) --------------------------------
  // Lane k propagates real basis vector k (k<16: Re(e_k), k>=16: Im(e_{k-16}))
  // through layer i's fixed-angle entangler; row k of U^T = resulting vector.
  if (wv == 0) {
    for (int i = 0; i < 3; ++i) {
      float sr[16], si[16];
#pragma unroll
      for (int n = 0; n < 16; ++n) {
        sr[n] = (lane == n) ? 1.f : 0.f;
        si[n] = (lane == (n + 16)) ? 1.f : 0.f;
      }
#pragma unroll
      for (int l = 0; l < 2; ++l) {
#pragma unroll
        for (int w = 0; w < 4; ++w) {
          float s, c;
          __sincosf(0.5f * theta[(i * 2 + l) * 4 + w], &s, &c);
#pragma unroll
          for (int n = 0; n < 16; ++n) {
            if ((n & (1 << w)) == 0) {
              const int m = n | (1 << w);
              float ar = sr[n], ai = si[n], br = sr[m], bi = si[m];
              sr[n] = c * ar + s * bi;  si[n] = c * ai - s * br;
              sr[m] = c * br + s * ai;  si[m] = c * bi - s * ar;
            }
          }
        }
#pragma unroll
        for (int cw = 0; cw < 4; ++cw) {
          const int tw = (cw + 1) & 3;
#pragma unroll
          for (int n = 0; n < 16; ++n) {
            if ((n & (1 << cw)) && !(n & (1 << tw))) {
              const int m = n | (1 << tw);
              float tr = sr[n]; sr[n] = sr[m]; sr[m] = tr;
              float ti = si[n]; si[n] = si[m]; si[m] = ti;
            }
          }
        }
      }
#pragma unroll
      for (int ng = 0; ng < 32; ++ng)
        lds_UT[i][lane * 32 + ng] = (ng < 16) ? sr[ng] : si[ng - 16];
    }
  }
  __syncthreads();

  const int tile = blockIdx.x * WAVES_PER_BLOCK + wv;
  const int r0 = tile * 16;
  if (r0 >= B) return;

  float* myh  = lds_h[wv];
  v2f*   mytr = lds_trig[wv];
  float* myS  = lds_S[wv];

#pragma unroll
  for (int layer = 0; layer < 3; ++layer) {
    // ---- dense + tanh -> angles (WMMA C layout; cols >=4 are exact zeros) ---
    v8f acc = {0.f, 0.f, 0.f, 0.f, 0.f, 0.f, 0.f, 0.f};
    if (layer == 0) {
      int rg = r0 + n16; if (rg >= B) rg = B - 1;
#pragma unroll
      for (int c = 0; c < 4; ++c) {
        const int k0 = 4 * c + 2 * hh;
        v2f a; a.x = x[(size_t)rg * 16 + k0]; a.y = x[(size_t)rg * 16 + k0 + 1];
        v2f b; b.x = lds_Wp[k0 * 16 + n16];   b.y = lds_Wp[(k0 + 1) * 16 + n16];
        acc = wmma4(a, b, acc);
      }
    } else {
      const int k0 = 2 * hh;
      const float* Wl = lds_Wp + ((layer == 1) ? 256 : 320);
      v2f a; a.x = myh[n16 * 16 + k0]; a.y = myh[n16 * 16 + k0 + 1];
      v2f b; b.x = Wl[k0 * 16 + n16];  b.y = Wl[(k0 + 1) * 16 + n16];
      acc = wmma4(a, b, acc);
    }
    const float bias = lds_bp[layer * 16 + n16];
    float ang[8];
#pragma unroll
    for (int j = 0; j < 8; ++j) ang[j] = tanh_fast(acc[j] + bias);

    // stage angles: row = j + 8*hh, col = n16 (all lanes, unconditional)
#pragma unroll
    for (int j = 0; j < 8; ++j) myh[(j + 8 * hh) * 16 + n16] = ang[j];

    // (cos,sin) once per (wire,row): lane -> row n16, wires {2hh, 2hh+1}
    // |angle| <= 1, so native sincos is safe
#pragma unroll
    for (int wq = 0; wq < 2; ++wq) {
      const int w = 2 * hh + wq;
      float s, c;
      __sincosf(0.5f * myh[n16 * 16 + w], &s, &c);
      v2f cs; cs.x = c; cs.y = s;
      mytr[w * 16 + n16] = cs;
    }

    // ---- circuit state in C layout: lane = component, VGPR j = row j/j+8 ----
    float re[8], im[8];
#pragma unroll
    for (int j = 0; j < 8; ++j) { re[j] = (n16 == 0) ? 1.f : 0.f; im[j] = 0.f; }

    // embedding RX: new.re = c*re + s*partner.im ; new.im = c*im - s*partner.re
#pragma unroll
    for (int w = 0; w < 4; ++w) {
#pragma unroll
      for (int j = 0; j < 8; ++j) {
        v2f cs = mytr[w * 16 + (j + 8 * hh)];
        float pr = shflx(re[j], 1 << w);
        float pi = shflx(im[j], 1 << w);
        float nr = cs.x * re[j] + cs.y * pi;
        float ni = cs.x * im[j] - cs.y * pr;
        re[j] = nr; im[j] = ni;
      }
    }

    // ---- entangler: out(16x32) = state(16x32) @ U^T via WMMA f32 16x16x4 ----
#pragma unroll
    for (int j = 0; j < 8; ++j) {
      myS[(j + 8 * hh) * 32 + n16]      = re[j];
      myS[(j + 8 * hh) * 32 + 16 + n16] = im[j];
    }
    const float* UT = lds_UT[layer];
    v8f nre = {0.f, 0.f, 0.f, 0.f, 0.f, 0.f, 0.f, 0.f};
    v8f nim = {0.f, 0.f, 0.f, 0.f, 0.f, 0.f, 0.f, 0.f};
#pragma unroll
    for (int c = 0; c < 8; ++c) {
      const int k0 = 4 * c + 2 * hh;
      v2f a;  a.x  = myS[n16 * 32 + k0];      a.y  = myS[n16 * 32 + k0 + 1];
      v2f br; br.x = UT[k0 * 32 + n16];       br.y = UT[(k0 + 1) * 32 + n16];
      v2f bi; bi.x = UT[k0 * 32 + 16 + n16];  bi.y = UT[(k0 + 1) * 32 + 16 + n16];
      nre = wmma4(a, br, nre);
      nim = wmma4(a, bi, nim);
    }

    // ---- Z expvals: Walsh-Hadamard over component lanes; ev_w at lane 2^w ---
#pragma unroll
    for (int j = 0; j < 8; ++j) {
      float p = nre[j] * nre[j] + nim[j] * nim[j];
#pragma unroll
      for (int st = 0; st < 4; ++st) {
        const int mask = 1 << st;
        float t = shflx(p, mask);
        p = (lane & mask) ? (t - p) : (t + p);
      }
      int w = -1;
      if (n16 == 1) w = 0; else if (n16 == 2) w = 1;
      else if (n16 == 4) w = 2; else if (n16 == 8) w = 3;
      if (w >= 0) {
        const int row = j + 8 * hh;
        if (layer < 2) {
          myh[row * 16 + w] = p;                 // next layer's dense input
        } else if (r0 + row < B) {
          out[(size_t)(r0 + row) * 4 + w] = p;   // final output
        }
      }
    }
  }
}

extern "C" void kernel_launch(void* const* d_in, const int* in_sizes, int n_in,
                              void* d_out, int out_size, void* d_ws,
                              size_t ws_size, hipStream_t stream) {
  const float* x     = (const float*)d_in[0];
  const float* theta = (const float*)d_in[1];
  const float* W0    = (const float*)d_in[2];
  const float* b0    = (const float*)d_in[3];
  const float* W1    = (const float*)d_in[4];
  const float* b1    = (const float*)d_in[5];
  const float* W2    = (const float*)d_in[6];
  const float* b2    = (const float*)d_in[7];
  float* out = (float*)d_out;

  const int B = in_sizes[0] / 16;
  const int tiles = (B + 15) / 16;
  const int blocks = (tiles + WAVES_PER_BLOCK - 1) / WAVES_PER_BLOCK;
  hqnn_kernel<<<blocks, THREADS, 0, stream>>>(x, theta, W0, b0, W1, b1, W2, b2,
                                              out, B);
}